// SurgiATM_4028679324355
// MI455X (gfx1250) — compile-verified
//
#include <hip/hip_runtime.h>

// Problem dims (fixed by the reference): B=16, C=3, H=W=512, fp32.
#define BN   16
#define CN   3
#define HN   512
#define WN   512
#define HWN  (HN * WN)        // 262144
#define CHWN (CN * HWN)       // 786432
#define NTOT (BN * CHWN)      // 12582912 elements per output tensor
#define RAD  7
#define WZ   15               // 15x15 erosion window
#define TY   16               // rows per block in vertical pass (= WZ+1, required by the
                              // Gil-Werman split below)

// ---------------------------------------------------------------------------
// CDNA5 async global->LDS copy (tracked with ASYNCcnt). Per-lane global
// address, per-lane LDS byte address (low 32 bits of the generic shared
// pointer are the LDS byte offset on gfx12xx apertures).
// ---------------------------------------------------------------------------
__device__ __forceinline__ void async_ld_b32(const void* gaddr, void* ldsaddr) {
    unsigned           l = (unsigned)(size_t)ldsaddr;
    unsigned long long g = (unsigned long long)gaddr;
    asm volatile("global_load_async_to_lds_b32 %0, %1, off"
                 :: "v"(l), "v"(g)
                 : "memory");
}

__device__ __forceinline__ void wait_async_all() {
    asm volatile("s_wait_asynccnt 0" ::: "memory");
}

__device__ __forceinline__ int clampi(int v, int lo, int hi) {
    return v < lo ? lo : (v > hi ? hi : v);
}

// ---------------------------------------------------------------------------
// Kernel A: channel-min + horizontal 15-tap min (replicate edges via clamp).
// One block = 256 output columns of one (b, y) row. Writes hmin[B,H,W].
// ---------------------------------------------------------------------------
__global__ __launch_bounds__(256) void SurgiATM_hmin_kernel(
    const float* __restrict__ smoky, float* __restrict__ hmin)
{
    __shared__ float s_cmin[256 + 2 * RAD];   // 270 floats

    const int tid = threadIdx.x;
    const int x0  = blockIdx.x * 256;
    const int y   = blockIdx.y;
    const int b   = blockIdx.z;

    const float* row = smoky + (size_t)b * CHWN + (size_t)y * WN;

    for (int i = tid; i < 256 + 2 * RAD; i += 256) {
        const int gx = clampi(x0 - RAD + i, 0, WN - 1);
        const float c0 = row[gx];
        const float c1 = row[HWN + gx];
        const float c2 = row[2 * HWN + gx];
        s_cmin[i] = fminf(c0, fminf(c1, c2));
    }
    __syncthreads();

    float m = s_cmin[tid];
#pragma unroll
    for (int k = 1; k < WZ; ++k) m = fminf(m, s_cmin[tid + k]);

    hmin[(size_t)b * HWN + (size_t)y * WN + (x0 + tid)] = m;
}

// ---------------------------------------------------------------------------
// Kernel B: vertical 15-tap min (async LDS staging) + fused elementwise.
// One block = 256 columns x TY rows of one image b.
//   dc  = min over 15 rows of hmin (replicate edges via clamp)
//   dct = (dc + 0.1) / 1.1
//   out0 = smoky - dct*model ; out1 = model ; out2 = dct*model
// ---------------------------------------------------------------------------
__global__ __launch_bounds__(256) void SurgiATM_fuse_kernel(
    const float* __restrict__ smoky, const float* __restrict__ model,
    const float* __restrict__ hmin, float* __restrict__ out)
{
    __shared__ float s_h[TY + 2 * RAD][256];  // 30 x 256 floats = 30 KB

    const int tid = threadIdx.x;
    const int x   = blockIdx.x * 256 + tid;
    const int y0  = blockIdx.y * TY;
    const int b   = blockIdx.z;

    // Stage 30 halo rows of this column strip into LDS via the async path.
    const float* hb = hmin + (size_t)b * HWN;
#pragma unroll
    for (int j = 0; j < TY + 2 * RAD; ++j) {
        const int gy = clampi(y0 - RAD + j, 0, HN - 1);
        async_ld_b32(hb + (size_t)gy * WN + x, &s_h[j][tid]);
    }
    wait_async_all();
    __syncthreads();

    // Pull this thread's column into registers (constant-indexed, fully SSA).
    float col[TY + 2 * RAD];
#pragma unroll
    for (int j = 0; j < TY + 2 * RAD; ++j) col[j] = s_h[j][tid];

    // Gil-Werman sliding min: suffix mins of col[0..14], prefix mins of
    // col[15..29]; window(i) = [i, i+14] = min(S[i], P[i-1]).
    float S[WZ], P[WZ];
    S[WZ - 1] = col[WZ - 1];
#pragma unroll
    for (int j = WZ - 2; j >= 0; --j) S[j] = fminf(col[j], S[j + 1]);
    P[0] = col[WZ];
#pragma unroll
    for (int j = 1; j < WZ; ++j) P[j] = fminf(P[j - 1], col[WZ + j]);

    float dc[TY];
    dc[0] = S[0];
#pragma unroll
    for (int i = 1; i < TY - 1; ++i) dc[i] = fminf(S[i], P[i - 1]);
    dc[TY - 1] = P[TY - 2];

    float* o0 = out;
    float* o1 = out + (size_t)NTOT;
    float* o2 = out + 2 * (size_t)NTOT;

#pragma unroll
    for (int i = 0; i < TY; ++i) {
        const float dct   = (dc[i] + 0.1f) * (1.0f / 1.1f);
        const size_t rbase = (size_t)b * CHWN + (size_t)(y0 + i) * WN + x;
#pragma unroll
        for (int c = 0; c < CN; ++c) {
            const size_t idx = rbase + (size_t)c * HWN;
            const float m   = model[idx];
            const float sdc = dct * m;
            const float pc  = smoky[idx] - sdc;
            // Write-once outputs: non-temporal so 151 MB of stores don't
            // evict smoky/model/hmin from the 192 MB L2.
            __builtin_nontemporal_store(pc,  &o0[idx]);
            __builtin_nontemporal_store(m,   &o1[idx]);
            __builtin_nontemporal_store(sdc, &o2[idx]);
        }
    }
}

// ---------------------------------------------------------------------------
extern "C" void kernel_launch(void* const* d_in, const int* in_sizes, int n_in,
                              void* d_out, int out_size, void* d_ws, size_t ws_size,
                              hipStream_t stream) {
    (void)in_sizes; (void)n_in; (void)out_size; (void)ws_size;

    const float* smoky = (const float*)d_in[0];
    const float* model = (const float*)d_in[1];
    float*       out   = (float*)d_out;
    float*       hmin  = (float*)d_ws;       // B*H*W fp32 = 16.8 MB scratch

    dim3 blk(256, 1, 1);

    dim3 gA(WN / 256, HN, BN);               // (2, 512, 16)
    SurgiATM_hmin_kernel<<<gA, blk, 0, stream>>>(smoky, hmin);

    dim3 gB(WN / 256, HN / TY, BN);          // (2, 32, 16)
    SurgiATM_fuse_kernel<<<gB, blk, 0, stream>>>(smoky, model, hmin, out);
}